// GNN4_17471926960718
// MI455X (gfx1250) — compile-verified
//
#include <hip/hip_runtime.h>

#define N_DRUG 846
#define DIM 256
#define KNB 64
#define BN_EPS 1e-5f

typedef __attribute__((ext_vector_type(2))) float v2f;
typedef __attribute__((ext_vector_type(8))) float v8f;

// Async copy of 16 contiguous bytes global -> LDS (CDNA5, ASYNCcnt-tracked).
__device__ __forceinline__ void async_copy_b128(unsigned lds_off, const void* gptr) {
    unsigned long long ga = (unsigned long long)(uintptr_t)gptr;
    asm volatile("global_load_async_to_lds_b128 %0, %1, off"
                 :
                 : "v"(lds_off), "v"(ga)
                 : "memory");
}
__device__ __forceinline__ void async_wait0() {
    asm volatile("s_wait_asynccnt 0" ::: "memory");
}
// LDS byte-offset of a __shared__ pointer (addrspace(3) is 32-bit offset space).
__device__ __forceinline__ unsigned lds_offset(const void* p) {
    return (unsigned)(uintptr_t)p;
}

// ---------------------------------------------------------------------------
// kernel 0: sb2[k] = sum_e b2[k,e]   (drug-independent, trivial)
// ---------------------------------------------------------------------------
__global__ void sb2_kernel(const float* __restrict__ b2, float* __restrict__ sb2) {
    int k = threadIdx.x;
    if (k < KNB) {
        float s = 0.f;
        for (int e = 0; e < DIM; ++e) s += b2[k * DIM + e];
        sb2[k] = s;
    }
}

// ---------------------------------------------------------------------------
// kernel 1: one workgroup per drug. Fused:
//   drug_rel = drug_emb * rel_emb                       (built chunkwise in LDS)
//   H        = relu(drug_rel @ W1[n] + b1)              (v_wmma_f32_16x16x4_f32)
//   logits   = H @ rowsum(W2[n]) + rowsum(b2)           (layer-2 GEMM collapsed)
//   score    = softmax(logits)
//   went     = score @ ent_emb  ;  z = relu([went,drug] @ lin_w + lin_b) -> ws
// W slabs staged with global_load_async_to_lds_b128 (ASYNCcnt).
// ---------------------------------------------------------------------------
__global__ __launch_bounds__(256)
void drug_kernel(const int* __restrict__ drug_name,
                 const int* __restrict__ adj_tail,
                 const int* __restrict__ adj_rel,
                 const float* __restrict__ drug_table,
                 const float* __restrict__ rela_table,
                 const float* __restrict__ ent_table,
                 const float* __restrict__ W1,
                 const float* __restrict__ b1,
                 const float* __restrict__ W2,
                 const float* __restrict__ lin_w,
                 const float* __restrict__ lin_b,
                 const float* __restrict__ sb2g,
                 float* __restrict__ zbuf) {
    // LDS (~48 KB)
    __shared__ float sAc[64 * 36];      // drug_rel K-chunk, padded stride 36
    __shared__ float sW[32 * 264];      // W slab, padded stride 264
    __shared__ float s2[DIM];           // rowsum(W2[n])
    __shared__ float sDrug[DIM];
    __shared__ float sDE[2 * DIM];      // [weighted_ent, drug_emb]
    __shared__ float sLogit[KNB];
    __shared__ float sScore[KNB];
    __shared__ float sRed[2];
    __shared__ int   sRel[KNB];
    __shared__ int   sTail[KNB];

    const int t      = threadIdx.x;
    const int n      = blockIdx.x;
    const int lane   = t & 31;
    const int wave   = t >> 5;          // 8 waves
    const int mstrip = wave >> 1;       // 4 M-strips of 16 rows
    const int nhalf  = wave & 1;        // column half (8 N-tiles each)
    const int hi     = lane >> 4;       // lane half (K / M offset)
    const int l15    = lane & 15;

    // ---- phase 0: gather per-drug metadata -------------------------------
    sDrug[t] = drug_table[(size_t)drug_name[n] * DIM + t];
    if (t < KNB) {
        sRel[t]   = adj_rel[n * KNB + t];
        sTail[t]  = adj_tail[n * KNB + t];
        sLogit[t] = sb2g[t];            // logits pre-seeded with rowsum(b2)
    }
    s2[t] = 0.f;
    __syncthreads();

    const float* W1n = W1 + (size_t)n * DIM * DIM;
    const float* W2n = W2 + (size_t)n * DIM * DIM;
    const unsigned sW_base = lds_offset(sW);

    // slab copy: 32x256 floats = 2048 b128 chunks -> 8 per thread.
    // chunk flat id = i*256 + t : row r = flat>>6 (0..31), quad q = flat&63.
    int srow[8]; unsigned sldso[8]; size_t sgoff[8];
    #pragma unroll
    for (int i = 0; i < 8; ++i) {
        int flat = i * 256 + t;
        int r = flat >> 6, q = flat & 63;
        srow[i]  = r;
        sldso[i] = sW_base + (unsigned)(r * 264 + q * 4) * 4u;
        sgoff[i] = (size_t)r * DIM + (size_t)q * 4;
    }

    v8f acc[8];
    #pragma unroll
    for (int i = 0; i < 8; ++i)
        #pragma unroll
        for (int j = 0; j < 8; ++j) acc[i][j] = 0.f;

    // ---- phase 1: GEMM1, K chunked by 32 ---------------------------------
    const int mrow = mstrip * 16 + l15;             // A-fragment row (M)
    for (int c = 0; c < 8; ++c) {
        const int kk0 = c * 32;
        __syncthreads();                // previous slab consumers done
        // async-stage W1 slab [32 x 256] into LDS (2048 x b128)
        const float* slab = W1n + (size_t)kk0 * DIM;
        #pragma unroll
        for (int i = 0; i < 8; ++i)
            async_copy_b128(sldso[i], slab + sgoff[i]);
        // build drug_rel chunk [64 x 32] while the DMA streams
        #pragma unroll
        for (int i = 0; i < 8; ++i) {
            int idx = i * 256 + t;
            int m = idx >> 5, cc = idx & 31;
            sAc[m * 36 + cc] =
                sDrug[kk0 + cc] * rela_table[(size_t)sRel[m] * DIM + kk0 + cc];
        }
        async_wait0();
        __syncthreads();

        // 8 K-steps x 8 N-tiles of v_wmma_f32_16x16x4_f32
        #pragma unroll
        for (int kk = 0; kk < 32; kk += 4) {
            v2f a = *(const v2f*)&sAc[mrow * 36 + kk + 2 * hi];   // A[m, kk+2hi..+1]
            #pragma unroll
            for (int j = 0; j < 8; ++j) {
                int col = (nhalf * 8 + j) * 16 + l15;
                v2f b;
                b.x = sW[(kk + 2 * hi)     * 264 + col];          // B[kk+2hi,   col]
                b.y = sW[(kk + 2 * hi + 1) * 264 + col];          // B[kk+2hi+1, col]
                acc[j] = __builtin_amdgcn_wmma_f32_16x16x4_f32(
                    false, a, false, b, (short)0, acc[j], false, false);
            }
        }
    }

    // ---- phase 2: s2 = rowsum(W2[n]) while streaming W2 ------------------
    for (int c = 0; c < 8; ++c) {
        __syncthreads();
        const float* slab = W2n + (size_t)(c * 32) * DIM;
        #pragma unroll
        for (int i = 0; i < 8; ++i)
            async_copy_b128(sldso[i], slab + sgoff[i]);
        async_wait0();
        __syncthreads();
        int r = t >> 3, p = t & 7;
        float s = 0.f;
        #pragma unroll
        for (int i = 0; i < 32; ++i) s += sW[r * 264 + p * 32 + i];
        atomicAdd(&s2[c * 32 + r], s);
    }
    __syncthreads();

    // ---- phase 3: bias + relu + dot with s2 -> logits --------------------
    float part[8];
    #pragma unroll
    for (int r = 0; r < 8; ++r) part[r] = 0.f;
    #pragma unroll
    for (int j = 0; j < 8; ++j) {
        int col = (nhalf * 8 + j) * 16 + l15;
        float s2c = s2[col];
        #pragma unroll
        for (int r = 0; r < 8; ++r) {                 // C layout: M = r + 8*hi
            int m = mstrip * 16 + r + 8 * hi;
            float v = acc[j][r] + b1[m * DIM + col];
            v = fmaxf(v, 0.f);
            part[r] += v * s2c;
        }
    }
    #pragma unroll
    for (int r = 0; r < 8; ++r)
        atomicAdd(&sLogit[mstrip * 16 + r + 8 * hi], part[r]);
    __syncthreads();

    // ---- phase 4: softmax over K=64 --------------------------------------
    if (t == 0) {
        float mx = sLogit[0];
        for (int k = 1; k < KNB; ++k) mx = fmaxf(mx, sLogit[k]);
        sRed[0] = mx;
    }
    __syncthreads();
    if (t < KNB) sScore[t] = __expf(sLogit[t] - sRed[0]);
    __syncthreads();
    if (t == 0) {
        float s = 0.f;
        for (int k = 0; k < KNB; ++k) s += sScore[k];
        sRed[1] = 1.f / s;
    }
    __syncthreads();

    // ---- phase 5: weighted neighbor aggregation + concat -----------------
    float inv = sRed[1];
    float w = 0.f;
    for (int k = 0; k < KNB; ++k)
        w += sScore[k] * ent_table[(size_t)sTail[k] * DIM + t];
    sDE[t]       = w * inv;
    sDE[DIM + t] = sDrug[t];
    __syncthreads();

    // ---- phase 6: Linear(512->256) + ReLU, z -> workspace ----------------
    float z = lin_b[t];
    for (int j = 0; j < 2 * DIM; ++j)
        z = fmaf(sDE[j], lin_w[j * DIM + t], z);
    z = fmaxf(z, 0.f);
    zbuf[(size_t)n * DIM + t] = z;
    (void)srow;
}

// ---------------------------------------------------------------------------
// kernel 2: BatchNorm1d over the batch (biased variance), one block per column
// ---------------------------------------------------------------------------
__global__ __launch_bounds__(256)
void bn_kernel(const float* __restrict__ zbuf,
               const float* __restrict__ gamma,
               const float* __restrict__ beta,
               float* __restrict__ out) {
    __shared__ float ss[256], sq[256];
    int e = blockIdx.x, t = threadIdx.x;
    float s = 0.f, q = 0.f;
    for (int r = t; r < N_DRUG; r += 256) {
        float v = zbuf[(size_t)r * DIM + e];
        s += v; q += v * v;
    }
    ss[t] = s; sq[t] = q;
    __syncthreads();
    for (int off = 128; off > 0; off >>= 1) {
        if (t < off) { ss[t] += ss[t + off]; sq[t] += sq[t + off]; }
        __syncthreads();
    }
    float mean = ss[0] / (float)N_DRUG;
    float var  = sq[0] / (float)N_DRUG - mean * mean;
    float g = gamma[e], b = beta[e];
    float invs = rsqrtf(var + BN_EPS);
    for (int r = t; r < N_DRUG; r += 256) {
        float v = zbuf[(size_t)r * DIM + e];
        out[(size_t)r * DIM + e] = g * (v - mean) * invs + b;
    }
}

// ---------------------------------------------------------------------------
extern "C" void kernel_launch(void* const* d_in, const int* in_sizes, int n_in,
                              void* d_out, int out_size, void* d_ws, size_t ws_size,
                              hipStream_t stream) {
    const int*   drug_name  = (const int*)d_in[0];
    const int*   adj_tail   = (const int*)d_in[1];
    const int*   adj_rel    = (const int*)d_in[2];
    const float* drug_table = (const float*)d_in[3];
    const float* rela_table = (const float*)d_in[4];
    const float* ent_table  = (const float*)d_in[5];
    const float* W1         = (const float*)d_in[6];
    const float* b1         = (const float*)d_in[7];
    const float* W2         = (const float*)d_in[8];
    const float* b2         = (const float*)d_in[9];
    const float* lin_w      = (const float*)d_in[10];
    const float* lin_b      = (const float*)d_in[11];
    const float* bn_gamma   = (const float*)d_in[12];
    const float* bn_beta    = (const float*)d_in[13];
    (void)in_sizes; (void)n_in; (void)out_size; (void)ws_size;

    float* wsf  = (float*)d_ws;
    float* sb2  = wsf;            // 64 floats
    float* zbuf = wsf + 64;       // N_DRUG*DIM floats

    sb2_kernel<<<1, 64, 0, stream>>>(b2, sb2);
    drug_kernel<<<N_DRUG, 256, 0, stream>>>(drug_name, adj_tail, adj_rel,
                                            drug_table, rela_table, ent_table,
                                            W1, b1, W2, lin_w, lin_b, sb2, zbuf);
    bn_kernel<<<DIM, 256, 0, stream>>>(zbuf, bn_gamma, bn_beta, (float*)d_out);
}